// Convolution_14173392077319
// MI455X (gfx1250) — compile-verified
//
#include <hip/hip_runtime.h>

typedef float v2f __attribute__((ext_vector_type(2)));
typedef float v8f __attribute__((ext_vector_type(8)));

#define N_NODES 10000
#define N_EDGES 320000

// ---------------- math helpers ----------------

__device__ __forceinline__ float hw_tanh(float x) {
#if __has_builtin(__builtin_amdgcn_tanhf)
    return __builtin_amdgcn_tanhf(x);
#else
    // gfx1250 hardware transcendental; v_nop covers the TRANS-op result hazard
    float r;
    asm volatile("v_tanh_f32 %0, %1\n\tv_nop" : "=v"(r) : "v"(x));
    return r;
#endif
}

__device__ __forceinline__ float gelu_tanh(float x) {
    // jax.nn.gelu approximate=True
    const float k = 0.7978845608028654f; // sqrt(2/pi)
    float x3 = x * x * x;
    return 0.5f * x * (1.0f + hw_tanh(k * (x + 0.044715f * x3)));
}

// ---------------- WMMA fragment helpers (fp32 16x16x4) ----------------
// A (16x4, MxK) layout per CDNA5 ISA: lanes 0-15 hold M=lane, K=k0,k0+1;
// lanes 16-31 hold M=lane-16, K=k0+2,k0+3.
// B (4x16, KxN) mirrored: lanes 0-15 hold N=lane, K=k0,k0+1; lanes 16-31 K=k0+2,k0+3.
// C/D (16x16): VGPR r, lanes 0-15 -> M=r, N=lane; lanes 16-31 -> M=r+8, N=lane-16.

__device__ __forceinline__ v2f lds_a_frag(const float* s, int stride, int lane, int k0) {
    int row = lane & 15;
    int kk  = k0 + ((lane >> 4) << 1);
    v2f a;
    a.x = s[row * stride + kk];
    a.y = s[row * stride + kk + 1];
    return a;
}

__device__ __forceinline__ v2f g_b_frag(const float* __restrict__ W, int ldw, int lane, int k0, int n0) {
    int col = n0 + (lane & 15);
    int kk  = k0 + ((lane >> 4) << 1);
    v2f b;
    b.x = W[kk * ldw + col];
    b.y = W[(kk + 1) * ldw + col];
    return b;
}

__device__ __forceinline__ v8f wmma4(v2f a, v2f b, v8f c) {
    return __builtin_amdgcn_wmma_f32_16x16x4_f32(false, a, false, b, (short)0, c, false, false);
}

// ---------------- kernel: zero accumulators ----------------

__global__ void zero_kernel(float* __restrict__ p, long n) {
    long i = (long)blockIdx.x * blockDim.x + threadIdx.x;
    if (i < n) p[i] = 0.0f;
}

// ---------------- kernel: per-node pre (sc_*, f_*) ----------------

__global__ __launch_bounds__(160) void node_pre_kernel(
    const float* __restrict__ node_input, const float* __restrict__ node_attr,
    const float* __restrict__ Wsc0, const float* __restrict__ Wsc1,
    const float* __restrict__ Wl1_0, const float* __restrict__ Wl1_1,
    float* __restrict__ f_s, float* __restrict__ f_v,
    float* __restrict__ sc_s, float* __restrict__ sc_v)
{
    int n = blockIdx.x;
    int t = threadIdx.x;
    __shared__ float xin[160];
    xin[t] = node_input[(long)n * 160 + t];
    __syncthreads();
    float a = node_attr[n];
    if (t < 64) {
        int j = t;
        float s1 = 0.f, s2 = 0.f;
        #pragma unroll 8
        for (int k = 0; k < 64; k++) {
            float x = xin[k];
            s1 += x * Wl1_0[k * 64 + j];
            s2 += x * Wsc0[k * 64 + j];
        }
        float sc = a * 0.125f; // 1/sqrt(64)
        f_s[(long)n * 64 + j]  = s1 * sc;
        sc_s[(long)n * 64 + j] = s2 * sc;
    } else {
        int idx = t - 64;            // 0..95 = w*3+c
        int w = idx / 3, c = idx - 3 * w;
        float v1 = 0.f, v2 = 0.f;
        #pragma unroll 8
        for (int u = 0; u < 32; u++) {
            float x = xin[64 + u * 3 + c];
            v1 += x * Wl1_1[u * 32 + w];
            v2 += x * Wsc1[u * 32 + w];
        }
        float sc = a * 0.17677669529663687f; // 1/sqrt(32)
        f_v[(long)n * 96 + idx]  = v1 * sc;
        sc_v[(long)n * 96 + idx] = v2 * sc;
    }
}

// ---------------- kernel: edge MLP (WMMA) + tensor product + scatter ----------------

__global__ __launch_bounds__(32) void edge_kernel(
    const float* __restrict__ esa, const float* __restrict__ eattr,
    const int* __restrict__ esrc, const int* __restrict__ edst,
    const float* __restrict__ Wfc1, const float* __restrict__ Wfc2,
    const float* __restrict__ Wfc3,
    const float* __restrict__ f_s, const float* __restrict__ f_v,
    float* __restrict__ ns0, float* __restrict__ ns1,
    float* __restrict__ nv2, float* __restrict__ nv3)
{
    __shared__ float h[16 * 65];    // activation staging, padded stride
    __shared__ float wb[16 * 193];  // w = [w0|w1|w2|w3] per edge, padded stride

    int lane = threadIdx.x;
    long E0  = (long)blockIdx.x * 16;
    int row  = lane & 15;
    int hi2  = (lane >> 4) << 1;    // K sub-offset: 0 or 2
    int mrow = (lane >> 4) << 3;    // C-store M offset: 0 or 8

    // ---- GEMM1: X(16x8) @ Wfc1(8x64), scale 1/sqrt(8), GELU -> h ----
    const float* xrow = esa + (E0 + row) * 8;
    v2f a0, a1;
    a0.x = xrow[hi2];     a0.y = xrow[hi2 + 1];
    a1.x = xrow[4 + hi2]; a1.y = xrow[5 + hi2];
    #pragma unroll
    for (int nt = 0; nt < 4; nt++) {
        v8f c = {};
        c = wmma4(a0, g_b_frag(Wfc1, 64, lane, 0, nt * 16), c);
        c = wmma4(a1, g_b_frag(Wfc1, 64, lane, 4, nt * 16), c);
        #pragma unroll
        for (int r = 0; r < 8; r++)
            h[(mrow + r) * 65 + nt * 16 + row] = gelu_tanh(c[r] * 0.35355339059327373f);
    }
    __syncthreads();

    // ---- GEMM2: h(16x64) @ Wfc2(64x64), scale 1/8, GELU -> h ----
    {
        v8f c0 = {}, c1 = {}, c2 = {}, c3 = {};
        #pragma unroll
        for (int ks = 0; ks < 16; ks++) {
            v2f a = lds_a_frag(h, 65, lane, ks * 4);
            c0 = wmma4(a, g_b_frag(Wfc2, 64, lane, ks * 4,  0), c0);
            c1 = wmma4(a, g_b_frag(Wfc2, 64, lane, ks * 4, 16), c1);
            c2 = wmma4(a, g_b_frag(Wfc2, 64, lane, ks * 4, 32), c2);
            c3 = wmma4(a, g_b_frag(Wfc2, 64, lane, ks * 4, 48), c3);
        }
        __syncthreads();
        #pragma unroll
        for (int r = 0; r < 8; r++) {
            h[(mrow + r) * 65 +  0 + row] = gelu_tanh(c0[r] * 0.125f);
            h[(mrow + r) * 65 + 16 + row] = gelu_tanh(c1[r] * 0.125f);
            h[(mrow + r) * 65 + 32 + row] = gelu_tanh(c2[r] * 0.125f);
            h[(mrow + r) * 65 + 48 + row] = gelu_tanh(c3[r] * 0.125f);
        }
    }
    __syncthreads();

    // ---- GEMM3: h(16x64) @ Wfc3(64x192), scale 1/8 -> wb ----
    #pragma unroll
    for (int g = 0; g < 3; g++) {
        v8f c0 = {}, c1 = {}, c2 = {}, c3 = {};
        int nb = g * 64;
        #pragma unroll
        for (int ks = 0; ks < 16; ks++) {
            v2f a = lds_a_frag(h, 65, lane, ks * 4);
            c0 = wmma4(a, g_b_frag(Wfc3, 192, lane, ks * 4, nb +  0), c0);
            c1 = wmma4(a, g_b_frag(Wfc3, 192, lane, ks * 4, nb + 16), c1);
            c2 = wmma4(a, g_b_frag(Wfc3, 192, lane, ks * 4, nb + 32), c2);
            c3 = wmma4(a, g_b_frag(Wfc3, 192, lane, ks * 4, nb + 48), c3);
        }
        #pragma unroll
        for (int r = 0; r < 8; r++) {
            wb[(mrow + r) * 193 + nb +  0 + row] = c0[r] * 0.125f;
            wb[(mrow + r) * 193 + nb + 16 + row] = c1[r] * 0.125f;
            wb[(mrow + r) * 193 + nb + 32 + row] = c2[r] * 0.125f;
            wb[(mrow + r) * 193 + nb + 48 + row] = c3[r] * 0.125f;
        }
    }
    __syncthreads();

    // ---- tensor product + atomic scatter: 2 lanes per edge ----
    int e = lane >> 1, hf = lane & 1;
    long E = E0 + e;
    int src = esrc[E];
    int dst = edst[E];
    float y0  = eattr[E * 4 + 0];
    float y1a = eattr[E * 4 + 1];
    float y1b = eattr[E * 4 + 2];
    float y1c = eattr[E * 4 + 3];
    const float* we = &wb[e * 193];
    const float* gs = &f_s[(long)src * 64];   // 256B-aligned rows
    const float* gv = &f_v[(long)src * 96];   // 384B-aligned rows
    float* p0 = &ns0[(long)dst * 64];
    float* p1 = &ns1[(long)dst * 32];
    float* p2 = &nv2[(long)dst * 192];
    float* p3 = &nv3[(long)dst * 96];

    // scalar outputs: j in [hf*32, hf*32+32), gather f_s via b128 loads
    #pragma unroll
    for (int ii = 0; ii < 8; ii++) {
        int j = hf * 32 + ii * 4;
        float4 g4 = *(const float4*)(gs + j);   // 16B-aligned
        float gq[4] = { g4.x, g4.y, g4.z, g4.w };
        #pragma unroll
        for (int q = 0; q < 4; q++) {
            int jj = j + q;
            float gsv = gq[q];
            float w0v = we[jj];
            float w1v = we[64 + jj];
            atomicAdd(&p0[jj], w0v * gsv * y0);
            float tt = w1v * gsv;
            atomicAdd(&p2[jj * 3 + 0], tt * y1a);
            atomicAdd(&p2[jj * 3 + 1], tt * y1b);
            atomicAdd(&p2[jj * 3 + 2], tt * y1c);
        }
    }
    // vector outputs: u in [hf*16, hf*16+16), gather f_v via b128 loads
    const float INV_SQRT3 = 0.5773502691896258f;
    #pragma unroll
    for (int ii = 0; ii < 4; ii++) {
        int ubase = hf * 16 + ii * 4;           // multiple of 4 -> 48B stride, 16B aligned
        const float* gp = gv + ubase * 3;
        float4 q0 = *(const float4*)(gp);
        float4 q1 = *(const float4*)(gp + 4);
        float4 q2 = *(const float4*)(gp + 8);
        float gg[12] = { q0.x, q0.y, q0.z, q0.w,
                         q1.x, q1.y, q1.z, q1.w,
                         q2.x, q2.y, q2.z, q2.w };
        #pragma unroll
        for (int q = 0; q < 4; q++) {
            int u = ubase + q;
            float g0 = gg[q * 3 + 0];
            float g1 = gg[q * 3 + 1];
            float g2 = gg[q * 3 + 2];
            float w2v = we[128 + u];
            float w3v = we[160 + u];
            atomicAdd(&p1[u], w3v * (g0 * y1a + g1 * y1b + g2 * y1c) * INV_SQRT3);
            atomicAdd(&p3[u * 3 + 0], w2v * g0 * y0);
            atomicAdd(&p3[u * 3 + 1], w2v * g1 * y0);
            atomicAdd(&p3[u * 3 + 2], w2v * g2 * y0);
        }
    }
}

// ---------------- kernel: per-node post (out GEMMs + gate + residual) ----------------

__global__ __launch_bounds__(160) void node_post_kernel(
    const float* __restrict__ node_attr,
    const float* __restrict__ Wl2_s, const float* __restrict__ Wl2_v,
    const float* __restrict__ Wa,
    const float* __restrict__ sc_s, const float* __restrict__ sc_v,
    const float* __restrict__ ns0, const float* __restrict__ ns1,
    const float* __restrict__ nv2, const float* __restrict__ nv3,
    float* __restrict__ out)
{
    int n = blockIdx.x;
    int t = threadIdx.x;
    __shared__ float ms[96];
    __shared__ float mv[288];
    const float inv = 0.17677669529663687f; // 1/sqrt(NUM_NEIGHBORS=32)

    if (t < 96)
        ms[t] = (t < 64 ? ns0[(long)n * 64 + t] : ns1[(long)n * 32 + (t - 64)]) * inv;
    for (int idx = t; idx < 288; idx += 160)
        mv[idx] = (idx < 192 ? nv2[(long)n * 192 + idx] : nv3[(long)n * 96 + (idx - 192)]) * inv;
    __syncthreads();

    float a   = node_attr[n];
    float s96 = a * 0.10206207261596575f; // a/sqrt(96)

    float alpha = 0.f;
    #pragma unroll 8
    for (int k = 0; k < 96; k++) alpha += ms[k] * Wa[k];
    alpha *= s96;

    if (t < 64) {
        int j = t;
        float o = 0.f;
        #pragma unroll 8
        for (int k = 0; k < 96; k++) o += ms[k] * Wl2_s[k * 64 + j];
        out[(long)n * 160 + j] = sc_s[(long)n * 64 + j] + alpha * (o * s96);
    } else {
        int idx = t - 64;
        int w = idx / 3, c = idx - 3 * w;
        float o = 0.f;
        #pragma unroll 8
        for (int u = 0; u < 96; u++) o += mv[u * 3 + c] * Wl2_v[u * 32 + w];
        out[(long)n * 160 + 64 + idx] = sc_v[(long)n * 96 + idx] + alpha * (o * s96);
    }
}

// ---------------- launch ----------------

extern "C" void kernel_launch(void* const* d_in, const int* in_sizes, int n_in,
                              void* d_out, int out_size, void* d_ws, size_t ws_size,
                              hipStream_t stream) {
    const float* node_input = (const float*)d_in[0];
    const float* node_attr  = (const float*)d_in[1];
    const float* edge_attr  = (const float*)d_in[2];
    const float* esa        = (const float*)d_in[3];
    const float* Wsc0       = (const float*)d_in[4];
    const float* Wsc1       = (const float*)d_in[5];
    const float* Wl1_0      = (const float*)d_in[6];
    const float* Wl1_1      = (const float*)d_in[7];
    const float* Wfc1       = (const float*)d_in[8];
    const float* Wfc2       = (const float*)d_in[9];
    const float* Wfc3       = (const float*)d_in[10];
    const float* Wl2_s      = (const float*)d_in[11];
    const float* Wl2_v      = (const float*)d_in[12];
    const float* Wa         = (const float*)d_in[13];
    const int*   edge_src   = (const int*)d_in[14];
    const int*   edge_dst   = (const int*)d_in[15];
    float* out = (float*)d_out;

    float* ws = (float*)d_ws;
    float* f_s  = ws;                    // 640000
    float* f_v  = ws + 640000;           // 960000
    float* sc_s = ws + 1600000;          // 640000
    float* sc_v = ws + 2240000;          // 960000
    float* ns0  = ws + 3200000;          // 640000
    float* ns1  = ws + 3840000;          // 320000
    float* nv2  = ws + 4160000;          // 1920000
    float* nv3  = ws + 6080000;          // 960000
    const long n_acc = 3840000;          // ns0..nv3 contiguous

    zero_kernel<<<(int)((n_acc + 255) / 256), 256, 0, stream>>>(ns0, n_acc);

    node_pre_kernel<<<N_NODES, 160, 0, stream>>>(
        node_input, node_attr, Wsc0, Wsc1, Wl1_0, Wl1_1, f_s, f_v, sc_s, sc_v);

    edge_kernel<<<N_EDGES / 16, 32, 0, stream>>>(
        esa, edge_attr, edge_src, edge_dst, Wfc1, Wfc2, Wfc3,
        f_s, f_v, ns0, ns1, nv2, nv3);

    node_post_kernel<<<N_NODES, 160, 0, stream>>>(
        node_attr, Wl2_s, Wl2_v, Wa, sc_s, sc_v, ns0, ns1, nv2, nv3, out);
}